// RoutingByAggreement_41154376630991
// MI455X (gfx1250) — compile-verified
//
#include <hip/hip_runtime.h>
#include <hip/hip_bf16.h>

// Capsule dynamic routing for MI455X (gfx1250).
// One workgroup per (batch, out_capsule). u_hat tile (1152x16 f32) staged to
// LDS once via TDM tensor_load_to_lds (row-padded to stride 17 dwords for
// bank-conflict-free access). 3 routing iterations run entirely from LDS.
// Agreement step a = U*v uses v_wmma_f32_16x16x4_f32.

#define N_IN   1152
#define N_OUT  10
#define CAPD   16
#define ROWS   (N_IN * N_OUT * CAPD)   // floats per batch
#define LSTRIDE 17                     // padded LDS row stride (dwords)
#define NTHREADS 256
#define NWAVES 8

typedef unsigned int u32x4 __attribute__((ext_vector_type(4)));
typedef int   i32x8 __attribute__((ext_vector_type(8)));
typedef int   i32x4 __attribute__((ext_vector_type(4)));
typedef float v2f   __attribute__((ext_vector_type(2)));
typedef float v8f   __attribute__((ext_vector_type(8)));

struct SMem {
    float tile[N_IN * LSTRIDE];     // 78336 B, must be first (TDM lds_addr = 0)
    float b[N_IN];                  // routing logits
    float part[NWAVES][CAPD + 1];   // per-wave partial sums (16 dims + expsum)
    float red[NWAVES];              // per-wave max reduction scratch
    float sv[CAPD];                 // s vector (pre-squash)
};

__device__ __forceinline__ float wave_sum(float v) {
    #pragma unroll
    for (int off = 16; off; off >>= 1) v += __shfl_down(v, off, 32);
    return v;
}
__device__ __forceinline__ float wave_max(float v) {
    #pragma unroll
    for (int off = 16; off; off >>= 1) v = fmaxf(v, __shfl_down(v, off, 32));
    return v;
}

__global__ __launch_bounds__(NTHREADS) void
caps_routing_kernel(const float* __restrict__ uhat, float* __restrict__ out) {
    __shared__ SMem sm;
    const int tid  = threadIdx.x;
    const int lane = tid & 31;
    const int wid  = tid >> 5;
    const int blk  = blockIdx.x;          // = batch * N_OUT + o
    const int bidx = blk / N_OUT;
    const int o    = blk - bidx * N_OUT;

    // ---------------- TDM load: u_hat[bidx, :, o, :] -> sm.tile -------------
    // 2D descriptor: tile_dim0=16 (contiguous dwords), tile_dim1=1152 rows,
    // tensor_dim0_stride=160 elem (= n_out*D). Pad 1 dword every 16 dwords so
    // LDS row stride = 17 dwords (conflict-free, 17 coprime with 64 banks).
    if (wid == 0) {
        unsigned long long ga = (unsigned long long)(const char*)
            (uhat + (size_t)bidx * ROWS + (size_t)o * CAPD);
        u32x4 g0;
        g0[0] = 1u;                                    // count=1 (valid D#)
        g0[1] = 0u;                                    // lds_addr = 0 (tile @ base)
        g0[2] = (unsigned)(ga & 0xFFFFFFFFu);          // global_addr[31:0]
        g0[3] = (unsigned)((ga >> 32) & 0x01FFFFFFu)   // global_addr[56:32]
              | (2u << 30);                            // type = 2 ("image")
        i32x8 g1;
        g1[0] = (2 << 16)        // data_size = 4 bytes
              | (1 << 20)        // pad_enable
              | (3 << 22);       // pad_interval: 16 dwords (pad_amount=0 -> 1 dword)
        g1[1] = (CAPD & 0xFFFF) << 16;                 // tensor_dim0 = 16
        g1[2] = (int)((unsigned)N_IN << 16);           // tensor_dim1 = 1152
        g1[3] = (CAPD & 0xFFFF) << 16;                 // tile_dim0 = 16
        g1[4] = N_IN;                                  // tile_dim1 = 1152 (tile_dim2=0)
        g1[5] = N_OUT * CAPD;                          // tensor_dim0_stride = 160
        g1[6] = 0;
        g1[7] = 0;
#if __has_builtin(__builtin_amdgcn_tensor_load_to_lds_d2)
        // Dedicated <=2D form: exactly group0 + group1 + cpol.
        __builtin_amdgcn_tensor_load_to_lds_d2(g0, g1, 0);
#else
        i32x4 gz4 = {0, 0, 0, 0};                      // groups 2/3 unused (2D)
        i32x8 gz8 = {0, 0, 0, 0, 0, 0, 0, 0};
        __builtin_amdgcn_tensor_load_to_lds(g0, g1, gz4, gz4, gz8, 0);
#endif
    }
    // init routing logits while the DMA is in flight
    for (int i = tid; i < N_IN; i += NTHREADS) sm.b[i] = 0.0f;
    if (wid == 0) __builtin_amdgcn_s_wait_tensorcnt(0);
    __syncthreads();

    // ---------------- 3 routing iterations ---------------------------------
    float coef = 0.0f;   // final squash coefficient (kept for output)
    #pragma unroll 1
    for (int r = 0; r < 3; ++r) {
        // (1) block max of b (softmax stabilizer)
        float m = -3.402823466e38f;
        for (int i = tid; i < N_IN; i += NTHREADS) m = fmaxf(m, sm.b[i]);
        m = wave_max(m);
        if (lane == 0) sm.red[wid] = m;
        __syncthreads();
        float bmax = sm.red[0];
        #pragma unroll
        for (int w = 1; w < NWAVES; ++w) bmax = fmaxf(bmax, sm.red[w]);

        // (2) fused exp + weighted sum:  s_d = sum_i e_i * U[i][d],  E = sum_i e_i
        float es = 0.0f;
        float ps[CAPD];
        #pragma unroll
        for (int d = 0; d < CAPD; ++d) ps[d] = 0.0f;
        for (int i = tid; i < N_IN; i += NTHREADS) {
            float e = expf(sm.b[i] - bmax);
            es += e;
            const float* row = &sm.tile[i * LSTRIDE];
            #pragma unroll
            for (int d = 0; d < CAPD; ++d) ps[d] += e * row[d];
        }
        es = wave_sum(es);
        #pragma unroll
        for (int d = 0; d < CAPD; ++d) ps[d] = wave_sum(ps[d]);
        if (lane == 0) {
            #pragma unroll
            for (int d = 0; d < CAPD; ++d) sm.part[wid][d] = ps[d];
            sm.part[wid][CAPD] = es;
        }
        __syncthreads();

        // (3) finalize s = (sum partials) / E
        float sumE = 0.0f;
        #pragma unroll
        for (int w = 0; w < NWAVES; ++w) sumE += sm.part[w][CAPD];
        if (tid < CAPD) {
            float s = 0.0f;
            #pragma unroll
            for (int w = 0; w < NWAVES; ++w) s += sm.part[w][tid];
            sm.sv[tid] = s / sumE;
        }
        __syncthreads();

        // (4) squash: v = ssq * s / ((1+ssq) * sqrt(ssq+eps))
        float ssq = 0.0f;
        #pragma unroll
        for (int d = 0; d < CAPD; ++d) { float s = sm.sv[d]; ssq += s * s; }
        coef = ssq / ((1.0f + ssq) * sqrtf(ssq + 1e-7f));

        if (r < 2) {
            float vreg[CAPD];
            #pragma unroll
            for (int d = 0; d < CAPD; ++d) vreg[d] = coef * sm.sv[d];

            // (5) agreement a_i = dot(U[i,:], v) via v_wmma_f32_16x16x4_f32.
            // A (16x4 f32): lanes 0-15 rows M with K={kc,kc+1}; lanes 16-31 same
            // rows with K={kc+2,kc+3}. B (4x16): v[k] broadcast across N.
            // D[m,n] = a[rowbase+m] for every n; read col 0 via lanes 0 / 16.
            const int mrow = lane & 15;
            const bool lowk = (lane < 16);
            #pragma unroll 1
            for (int g = wid; g < N_IN / 16; g += NWAVES) {
                const int rowbase = g * 16;
                const float* arow = &sm.tile[(rowbase + mrow) * LSTRIDE];
                v8f acc = {0.f, 0.f, 0.f, 0.f, 0.f, 0.f, 0.f, 0.f};
                #pragma unroll
                for (int kc = 0; kc < CAPD; kc += 4) {
                    v2f A, Bv;
                    if (lowk) {
                        A[0]  = arow[kc + 0]; A[1]  = arow[kc + 1];
                        Bv[0] = vreg[kc + 0]; Bv[1] = vreg[kc + 1];
                    } else {
                        A[0]  = arow[kc + 2]; A[1]  = arow[kc + 3];
                        Bv[0] = vreg[kc + 2]; Bv[1] = vreg[kc + 3];
                    }
                    acc = __builtin_amdgcn_wmma_f32_16x16x4_f32(
                        false, A, false, Bv, (short)0, acc, false, false);
                }
                // C/D layout: VGPR j = row M=j (lanes 0-15) / M=j+8 (lanes 16-31)
                if (lane == 0) {
                    #pragma unroll
                    for (int j = 0; j < 8; ++j) sm.b[rowbase + j] += acc[j];
                } else if (lane == 16) {
                    #pragma unroll
                    for (int j = 0; j < 8; ++j) sm.b[rowbase + 8 + j] += acc[j];
                }
            }
        }
        __syncthreads();
    }

    // ---------------- output: v = coef * s  -> out[blk*16 + d] -------------
    if (tid < CAPD) out[(size_t)blk * CAPD + tid] = coef * sm.sv[tid];
}

extern "C" void kernel_launch(void* const* d_in, const int* in_sizes, int n_in,
                              void* d_out, int out_size, void* d_ws, size_t ws_size,
                              hipStream_t stream) {
    const float* uhat = (const float*)d_in[0];
    float* out = (float*)d_out;
    const int B = in_sizes[0] / ROWS;          // 256
    dim3 grid(B * N_OUT), block(NTHREADS);
    caps_routing_kernel<<<grid, block, 0, stream>>>(uhat, out);
}